// ChildSumTreeLSTMEncoder_33921651704011
// MI455X (gfx1250) — compile-verified
//
#include <hip/hip_runtime.h>
#include <hip/hip_bf16.h>

// ---------------------------------------------------------------------------
// Child-Sum Tree-LSTM, fused per-level kernel using CDNA5 fp32 WMMA
// (v_wmma_f32_16x16x4_f32).  One block = 32 nodes, 16 waves (512 thr).
// LDS trimmed to ~150 KB so TWO workgroups fit in the 320 KB WGP (8 waves
// per SIMD32):  iou accumulates in-place into the pre tile via the WMMA
// C-operand, and h_sum reuses the dead x-tile region.  B fragments are
// loaded once per k-step and reused across row-tiles in registers.
// ---------------------------------------------------------------------------

typedef __attribute__((ext_vector_type(2))) float v2f;
typedef __attribute__((ext_vector_type(8))) float v8f;

#define LVLS    6
#define NNODES  65536
#define KCH     4
#define HDIM    128
#define MROWS   32            // nodes per block
#define NTHR    512           // 16 waves

// LDS row strides (padded to avoid 64-bank conflicts)
#define AX_LD   132   // x tile      32 x 128   (reused for h_sum)
#define CH_LD   132   // child_h    128 x 128   (reused for fpre)
#define PRE_LD  516   // pre tile    32 x 512   (gates accumulate in place)

__device__ __forceinline__ float sigmoidf_(float x) {
  return 1.0f / (1.0f + __expf(-x));
}

// A fragment (16x4 fp32) from LDS.  Per ISA 7.12.2:
//   lanes 0-15 : M=lane,    k = k0, k0+1
//   lanes 16-31: M=lane-16, k = k0+2, k0+3
__device__ __forceinline__ v2f lds_afrag(const float* A, int lda, int row0,
                                         int k0, int lane) {
  int m  = lane & 15;
  int kk = k0 + ((lane >> 4) << 1);
  const float* p = A + (row0 + m) * lda + kk;
  v2f a;
  a.x = p[0];
  a.y = p[1];
  return a;
}

// B fragment (4x16 fp32) streamed from global (weights are L2-resident).
__device__ __forceinline__ v2f glb_bfrag(const float* B, int ldb, int k0,
                                         int col0, int lane) {
  int n  = lane & 15;
  int kk = k0 + ((lane >> 4) << 1);
  v2f b;
  b.x = B[(size_t)kk * ldb + col0 + n];
  b.y = B[(size_t)(kk + 1) * ldb + col0 + n];
  return b;
}

// C/D fragment (16x16 fp32, 8 VGPRs): vgpr r -> lanes 0-15 M=r,
// lanes 16-31 M=r+8, N = lane&15.
__device__ __forceinline__ v8f lds_load_c(const float* C, int ldc, int row0,
                                          int col0, int lane) {
  int n  = lane & 15;
  int mo = (lane >> 4) << 3;
  v8f c;
#pragma unroll
  for (int r = 0; r < 8; ++r)
    c[r] = C[(row0 + mo + r) * ldc + col0 + n];
  return c;
}

__device__ __forceinline__ void lds_store_d(float* C, int ldc, int row0,
                                            int col0, v8f acc, int lane) {
  int n  = lane & 15;
  int mo = (lane >> 4) << 3;
#pragma unroll
  for (int r = 0; r < 8; ++r)
    C[(row0 + mo + r) * ldc + col0 + n] = acc[r];
}

__device__ __forceinline__ v8f wmma_f32(v2f a, v2f b, v8f c) {
  return __builtin_amdgcn_wmma_f32_16x16x4_f32(
      /*neg_a=*/false, a, /*neg_b=*/false, b,
      /*c_mod=*/(short)0, c, /*reuse_a=*/false, /*reuse_b=*/false);
}

template <bool HAS_CHILDREN>
__global__ __launch_bounds__(NTHR)
void tree_lstm_level(const int* __restrict__ tokens_l,
                     const int* __restrict__ tmask_l,
                     const int* __restrict__ child_ix_l,
                     const int* __restrict__ child_mask_l,
                     const float* __restrict__ emb,
                     const float* __restrict__ Wx,      // [128, 512]
                     const float* __restrict__ bias,    // [512]
                     const float* __restrict__ U_iou,   // [128, 384]
                     const float* __restrict__ Uf,      // [128, 128]
                     const float* __restrict__ h_prev,  // [N, 128]
                     const float* __restrict__ c_prev,  // [N, 128]
                     float* __restrict__ h_out,         // [N, 128]
                     float* __restrict__ c_out)         // [N, 128]
{
  extern __shared__ float lds[];
  float* sAx  = lds;                        // 32 x AX_LD  (x -> h_sum)
  float* sCH  = sAx + MROWS * AX_LD;        // 128 x CH_LD (child_h -> fpre)
  float* sPre = sCH + 4 * MROWS * CH_LD;    // 32 x PRE_LD (pre -> pre+iou)

  const int tid   = threadIdx.x;
  const int lane  = tid & 31;
  const int wave  = tid >> 5;               // 0..15
  const int node0 = blockIdx.x * MROWS;

  // ---- Phase 1: masked embedding gather -> sAx (32 x 128) -----------------
  {
    int row  = tid >> 4;                    // 0..31
    int cb   = (tid & 15) * 8;              // 0..120
    int node = node0 + row;
    int tok  = tokens_l[node];
    float m  = (float)tmask_l[node];
    const float4* e4 = (const float4*)(emb + (size_t)tok * HDIM + cb);
    float4* d4 = (float4*)(sAx + row * AX_LD + cb);
#pragma unroll
    for (int j = 0; j < 2; ++j) {
      float4 v = e4[j];
      v.x *= m; v.y *= m; v.z *= m; v.w *= m;
      d4[j] = v;
    }
  }

  // ---- Phase 2: masked child_h gather -> sCH (128 x 128) ------------------
  // Also prefetch the matching c_prev cachelines for the combine phase.
  if constexpr (HAS_CHILDREN) {
    int r    = tid >> 2;                    // 0..127 = node*4 + k
    int cb   = (tid & 3) * 32;              // 0,32,64,96
    int node = node0 + (r >> 2);
    int k    = r & 3;
    int ci   = child_ix_l[node * KCH + k];
    float m  = (float)child_mask_l[node * KCH + k];
    const float4* h4 = (const float4*)(h_prev + (size_t)ci * HDIM + cb);
    float4* d4 = (float4*)(sCH + r * CH_LD + cb);
    __builtin_prefetch(c_prev + (size_t)ci * HDIM + cb, 0, 0);
#pragma unroll
    for (int j = 0; j < 8; ++j) {
      float4 v = h4[j];
      v.x *= m; v.y *= m; v.z *= m; v.w *= m;
      d4[j] = v;
    }
  }
  __syncthreads();

  // ---- Phase 3: pre = x @ Wx + b   (32x128 @ 128x512) ---------------------
  // 2 row-tiles x 32 col-tiles; wave w owns col strips {w, w+16}; one B-frag
  // load feeds both row-tiles.
  {
#pragma unroll
    for (int cs = 0; cs < 2; ++cs) {
      int col0 = (wave + cs * 16) * 16;
      float bv = bias[col0 + (lane & 15)];
      v8f acc0, acc1;
#pragma unroll
      for (int r = 0; r < 8; ++r) { acc0[r] = bv; acc1[r] = bv; }
#pragma unroll 4
      for (int k0 = 0; k0 < HDIM; k0 += 4) {
        v2f bf = glb_bfrag(Wx, 512, k0, col0, lane);
        v2f a0 = lds_afrag(sAx, AX_LD, 0, k0, lane);
        v2f a1 = lds_afrag(sAx, AX_LD, 16, k0, lane);
        acc0 = wmma_f32(a0, bf, acc0);
        acc1 = wmma_f32(a1, bf, acc1);
      }
      lds_store_d(sPre, PRE_LD, 0, col0, acc0, lane);
      lds_store_d(sPre, PRE_LD, 16, col0, acc1, lane);
    }
  }
  __syncthreads();   // sPre valid; sAx dead (x no longer needed)

  if constexpr (HAS_CHILDREN) {
    // ---- Phase 4: h_sum = sum_k child_h -> sAx (reused region) ------------
    {
      int row = tid >> 4;
      int cb  = (tid & 15) * 8;
      const float4* s0 = (const float4*)(sCH + (row * 4 + 0) * CH_LD + cb);
      const float4* s1 = (const float4*)(sCH + (row * 4 + 1) * CH_LD + cb);
      const float4* s2 = (const float4*)(sCH + (row * 4 + 2) * CH_LD + cb);
      const float4* s3 = (const float4*)(sCH + (row * 4 + 3) * CH_LD + cb);
      float4* d4 = (float4*)(sAx + row * AX_LD + cb);
#pragma unroll
      for (int j = 0; j < 2; ++j) {
        float4 a = s0[j], b = s1[j], c = s2[j], d = s3[j];
        float4 v;
        v.x = a.x + b.x + c.x + d.x;
        v.y = a.y + b.y + c.y + d.y;
        v.z = a.z + b.z + c.z + d.z;
        v.w = a.w + b.w + c.w + d.w;
        d4[j] = v;
      }
    }

    // ---- Phase 5: fpre = child_h @ Uf  (128x128 @ 128x128) ----------------
    // 8 row-tiles x 8 col-tiles; wave w owns col strip (w&7), row-tiles
    // {4*(w>>3) .. +3}; one B-frag load feeds 4 WMMAs.
    v8f facc[4];
    {
      int col0  = (wave & 7) * 16;
      int rbase = (wave >> 3) * 4;
#pragma unroll
      for (int j = 0; j < 4; ++j) facc[j] = {};
#pragma unroll 4
      for (int k0 = 0; k0 < HDIM; k0 += 4) {
        v2f bf = glb_bfrag(Uf, HDIM, k0, col0, lane);
#pragma unroll
        for (int j = 0; j < 4; ++j) {
          v2f af = lds_afrag(sCH, CH_LD, (rbase + j) * 16, k0, lane);
          facc[j] = wmma_f32(af, bf, facc[j]);
        }
      }
    }
    __syncthreads();   // all sCH reads done; h_sum in sAx visible

    // fpre overwrites the child_h region
    {
      int col0  = (wave & 7) * 16;
      int rbase = (wave >> 3) * 4;
#pragma unroll
      for (int j = 0; j < 4; ++j)
        lds_store_d(sCH, CH_LD, (rbase + j) * 16, col0, facc[j], lane);
    }

    // ---- Phase 6: iou accumulated in place: pre += h_sum @ U_iou ----------
    // Source col-tiles 0..23 map onto pre columns: i (0-127) -> xi (+0),
    // o (128-255) -> xo (+128), u (256-383) -> xu (+128).  Existing pre tile
    // is loaded as the WMMA C operand; each dest tile is RMW'd by one wave.
    {
      int scol = wave * 16;                        // 0..240
      int dcol = (scol < 128) ? scol : scol + 128;
      v8f acc0 = lds_load_c(sPre, PRE_LD, 0, dcol, lane);
      v8f acc1 = lds_load_c(sPre, PRE_LD, 16, dcol, lane);
#pragma unroll 4
      for (int k0 = 0; k0 < HDIM; k0 += 4) {
        v2f bf = glb_bfrag(U_iou, 384, k0, scol, lane);
        v2f a0 = lds_afrag(sAx, AX_LD, 0, k0, lane);
        v2f a1 = lds_afrag(sAx, AX_LD, 16, k0, lane);
        acc0 = wmma_f32(a0, bf, acc0);
        acc1 = wmma_f32(a1, bf, acc1);
      }
      lds_store_d(sPre, PRE_LD, 0, dcol, acc0, lane);
      lds_store_d(sPre, PRE_LD, 16, dcol, acc1, lane);
    }
    {
      int scol = (16 + (wave >> 1)) * 16;          // 256..368 ("u" block)
      int dcol = scol + 128;                       // -> xu columns
      int row0 = (wave & 1) * 16;
      v8f acc = lds_load_c(sPre, PRE_LD, row0, dcol, lane);
#pragma unroll 4
      for (int k0 = 0; k0 < HDIM; k0 += 4) {
        v2f bf = glb_bfrag(U_iou, 384, k0, scol, lane);
        v2f af = lds_afrag(sAx, AX_LD, row0, k0, lane);
        acc = wmma_f32(af, bf, acc);
      }
      lds_store_d(sPre, PRE_LD, row0, dcol, acc, lane);
    }
  }
  __syncthreads();

  // ---- Phase 7: gates + cell update, write h / c --------------------------
  // sPre now holds [xi+i | xf | xo+o | xu+u].  32x128 elems; 8 per thread.
#pragma unroll
  for (int i = 0; i < 8; ++i) {
    int e    = tid + i * NTHR;
    int row  = e >> 7;
    int hcol = e & 127;
    int node = node0 + row;

    float zi = sPre[row * PRE_LD + hcol];
    float xf = sPre[row * PRE_LD + 128 + hcol];
    float zo = sPre[row * PRE_LD + 256 + hcol];
    float zu = sPre[row * PRE_LD + 384 + hcol];

    float fc = 0.0f;
    if constexpr (HAS_CHILDREN) {
#pragma unroll
      for (int k = 0; k < KCH; ++k) {
        float fp = sCH[(row * 4 + k) * CH_LD + hcol];   // fpre
        int   ci = child_ix_l[node * KCH + k];
        float m  = (float)child_mask_l[node * KCH + k];
        float cc = c_prev[(size_t)ci * HDIM + hcol] * m;
        fc += sigmoidf_(xf + fp) * cc;
      }
    }
    float iv = sigmoidf_(zi);
    float ov = sigmoidf_(zo);
    float uv = tanhf(zu);
    float c  = iv * uv + fc;
    float h  = ov * tanhf(c);
    c_out[(size_t)node * HDIM + hcol] = c;
    h_out[(size_t)node * HDIM + hcol] = h;
  }
}

extern "C" void kernel_launch(void* const* d_in, const int* in_sizes, int n_in,
                              void* d_out, int out_size, void* d_ws, size_t ws_size,
                              hipStream_t stream) {
  (void)in_sizes; (void)n_in; (void)out_size; (void)ws_size;
  const int*   tokens     = (const int*)d_in[0];   // [L, N]
  const int*   token_mask = (const int*)d_in[1];   // [L, N]
  const int*   child_ix   = (const int*)d_in[2];   // [L-1, N, K]
  const int*   child_mask = (const int*)d_in[3];   // [L-1, N, K]
  const float* emb        = (const float*)d_in[4]; // [V, 128]
  const float* Wx         = (const float*)d_in[5]; // [128, 512]
  const float* bias       = (const float*)d_in[6]; // [512]
  const float* U_iou      = (const float*)d_in[7]; // [128, 384]
  const float* Uf         = (const float*)d_in[8]; // [128, 128]
  float*       out        = (float*)d_out;         // [L, N, 128]
  float*       cbuf       = (float*)d_ws;          // 2 x [N, 128] ping-pong

  const size_t lvl_elems = (size_t)NNODES * HDIM;
  const size_t lds_bytes =
      (MROWS * AX_LD + 4 * MROWS * CH_LD + MROWS * PRE_LD) * sizeof(float);
  // = 150,528 B -> two workgroups per 320 KB WGP

  dim3 grid(NNODES / MROWS), block(NTHR);

  for (int l = LVLS - 1; l >= 0; --l) {
    float*       c_cur  = cbuf + (size_t)(l & 1) * lvl_elems;
    const float* c_prev = cbuf + (size_t)((l + 1) & 1) * lvl_elems;
    float*       h_out  = out + (size_t)l * lvl_elems;
    const float* h_prev = out + (size_t)(l + 1) * lvl_elems;

    if (l == LVLS - 1) {
      tree_lstm_level<false><<<grid, block, lds_bytes, stream>>>(
          tokens + (size_t)l * NNODES, token_mask + (size_t)l * NNODES,
          nullptr, nullptr, emb, Wx, bias, U_iou, Uf,
          nullptr, nullptr, h_out, c_cur);
    } else {
      tree_lstm_level<true><<<grid, block, lds_bytes, stream>>>(
          tokens + (size_t)l * NNODES, token_mask + (size_t)l * NNODES,
          child_ix + (size_t)l * NNODES * KCH,
          child_mask + (size_t)l * NNODES * KCH,
          emb, Wx, bias, U_iou, Uf, h_prev, c_prev, h_out, c_cur);
    }
  }
}